// deltaSPHModule_47021301957202
// MI455X (gfx1250) — compile-verified
//
#include <hip/hip_runtime.h>
#include <stdint.h>

// delta-SPH constants (folded from reference, double-precision folded to f32)
#define SPH_INVH   0.5f                   // 1/H, H = 2.0
#define SPH_EPS    0.4f                   // H*H*0.1
#define SPH_CW     0.5570423008216338f    // 7/(pi*H*H) = 7/(4pi)
#define SPH_GRADC  (-2.785211504108169f)  // -20*C_W/H^2 = -5*C_W
#define SPH_DHC    4.852216120579415f     // DELTA*H*C0 = 0.1*2*24.2610806...

#define TILE_EDGES 2048
#define BLOCK      256
#define EPT        (TILE_EDGES / BLOCK)   // 8 edges per thread

__global__ __launch_bounds__(BLOCK) void sph_zero_kernel(float* __restrict__ out, int n) {
  int i = blockIdx.x * blockDim.x + threadIdx.x;
  if (i < n) out[i] = 0.0f;
}

__device__ __forceinline__ void flush_row(float* __restrict__ out, int r,
                                          float a0, float a1, float a2) {
  // non-returning hardware f32 atomics (global_atomic_add_f32), device scope
  __hip_atomic_fetch_add(&out[3 * r + 0], a0, __ATOMIC_RELAXED, __HIP_MEMORY_SCOPE_AGENT);
  __hip_atomic_fetch_add(&out[3 * r + 1], a1, __ATOMIC_RELAXED, __HIP_MEMORY_SCOPE_AGENT);
  __hip_atomic_fetch_add(&out[3 * r + 2], a2, __ATOMIC_RELAXED, __HIP_MEMORY_SCOPE_AGENT);
}

struct IState { float pix, piy, vix, viy, di; };

__device__ __forceinline__ void load_istate(const float2* __restrict__ pos,
                                            const float2* __restrict__ vel,
                                            const float*  __restrict__ dens,
                                            int i, IState& st) {
  const float2 pi = pos[i];
  const float2 vi = vel[i];
  st.pix = pi.x; st.piy = pi.y; st.vix = vi.x; st.viy = vi.y;
  st.di = dens[i];
}

__device__ __forceinline__ void accum_edge(const float2* __restrict__ pos,
                                           const float2* __restrict__ vel,
                                           const float*  __restrict__ mass,
                                           const float*  __restrict__ dens,
                                           const IState& st, int j,
                                           float& a0, float& a1, float& a2) {
  const float2 pj = pos[j];
  const float xx = st.pix - pj.x;
  const float xy = st.piy - pj.y;
  const float r2 = xx * xx + xy * xy;
  const float r  = sqrtf(r2 + 1e-12f);
  const float q  = fminf(r * SPH_INVH, 1.0f);
  const float omq  = 1.0f - q;
  const float omq2 = omq * omq;
  const float W  = SPH_CW * (omq2 * omq2) * (4.0f * q + 1.0f);
  const float gs = SPH_GRADC * (omq2 * omq);          // gradW = gs * xij

  const float  mj = mass[j];
  const float2 vj = vel[j];
  const float  dj = dens[j];

  const float vdotx  = (st.vix - vj.x) * xx + (st.viy - vj.y) * xy;
  const float c_cont = mj * (vdotx * gs);
  // psi.gradW = 2*(dj-di)/(r2+EPS) * r2 * gs ; V_j = mj/dj -> fold to ONE divide
  const float c_diff = ((2.0f * SPH_DHC) * mj * (dj - st.di) * (r2 * gs))
                       / (dj * (r2 + SPH_EPS));
  a0 += mj * W;
  a1 += c_cont + c_diff;
  a2 += c_diff;
}

__global__ __launch_bounds__(BLOCK) void sph_edge_kernel(
    const float2* __restrict__ pos, const float2* __restrict__ vel,
    const float*  __restrict__ mass, const float* __restrict__ dens,
    const int*    __restrict__ row,  const int*   __restrict__ col,
    float* __restrict__ out, long long E) {
  __shared__ __align__(16) int s_row[TILE_EDGES];
  __shared__ __align__(16) int s_col[TILE_EDGES];

  const int tid = threadIdx.x;
  const long long tileBase = (long long)blockIdx.x * TILE_EDGES;

  // --- Stage edge-index stream into LDS with CDNA5 async copies (ASYNCcnt) --
  // TILE_EDGES ints per array = 8KB = 512 x 16B chunks; each thread owns 2
  // chunks per array; no VGPR round-trip.
#pragma unroll
  for (int c = 0; c < TILE_EDGES / (BLOCK * 4); ++c) {
    const int chunk = tid + c * BLOCK;                 // 16-byte chunk id
    const long long eStart = tileBase + (long long)chunk * 4;
    if (eStart < E) {
      uint32_t ldsR = (uint32_t)(uintptr_t)&s_row[chunk * 4];
      uint32_t ldsC = (uint32_t)(uintptr_t)&s_col[chunk * 4];
      const int* gR = row + eStart;
      const int* gC = col + eStart;
      asm volatile("global_load_async_to_lds_b128 %0, %1, off"
                   :: "v"(ldsR), "v"(gR) : "memory");
      asm volatile("global_load_async_to_lds_b128 %0, %1, off"
                   :: "v"(ldsC), "v"(gC) : "memory");
    }
  }
  asm volatile("s_wait_asynccnt 0" ::: "memory");
  __syncthreads();

  // --- Run-length segmented accumulation (row is globally sorted) -----------
  const int lbase = tid * EPT;
  const long long eBase = tileBase + lbase;

  int    cur = -1;
  IState st  = {};
  float  a0 = 0.f, a1 = 0.f, a2 = 0.f;

  if (tileBase + TILE_EDGES <= E) {
    // Fast path: full tile, no per-edge bounds checks; b128 LDS index reads.
    int ri[EPT], ci[EPT];
    *(int4*)&ri[0] = *(const int4*)&s_row[lbase + 0];
    *(int4*)&ri[4] = *(const int4*)&s_row[lbase + 4];
    *(int4*)&ci[0] = *(const int4*)&s_col[lbase + 0];
    *(int4*)&ci[4] = *(const int4*)&s_col[lbase + 4];
#pragma unroll
    for (int k = 0; k < EPT; ++k) {
      const int i = ri[k];
      if (i != cur) {
        if (cur >= 0) flush_row(out, cur, a0, a1, a2);
        cur = i; a0 = a1 = a2 = 0.f;
        load_istate(pos, vel, dens, i, st);
      }
      accum_edge(pos, vel, mass, dens, st, ci[k], a0, a1, a2);
    }
  } else {
    // Tail path: guarded per edge (only the last block takes this).
    for (int k = 0; k < EPT; ++k) {
      if (eBase + k >= E) break;
      const int i = s_row[lbase + k];
      const int j = s_col[lbase + k];
      if (i != cur) {
        if (cur >= 0) flush_row(out, cur, a0, a1, a2);
        cur = i; a0 = a1 = a2 = 0.f;
        load_istate(pos, vel, dens, i, st);
      }
      accum_edge(pos, vel, mass, dens, st, j, a0, a1, a2);
    }
  }
  if (cur >= 0) flush_row(out, cur, a0, a1, a2);
}

extern "C" void kernel_launch(void* const* d_in, const int* in_sizes, int n_in,
                              void* d_out, int out_size, void* d_ws, size_t ws_size,
                              hipStream_t stream) {
  const float2* pos  = (const float2*)d_in[0];
  const float2* vel  = (const float2*)d_in[1];
  const float*  mass = (const float*) d_in[2];
  const float*  dens = (const float*) d_in[3];
  const int*    row  = (const int*)   d_in[4];
  const int*    col  = (const int*)   d_in[5];
  float* out = (float*)d_out;
  const long long E = (long long)in_sizes[4];

  // output is accumulated with atomics -> must start from zero every call
  sph_zero_kernel<<<(out_size + BLOCK - 1) / BLOCK, BLOCK, 0, stream>>>(out, out_size);

  const int nblocks = (int)((E + TILE_EDGES - 1) / TILE_EDGES);
  sph_edge_kernel<<<nblocks, BLOCK, 0, stream>>>(pos, vel, mass, dens, row, col, out, E);
}